// OrderInvariantCapsuleLikelihood_27908697490038
// MI455X (gfx1250) — compile-verified
//
#include <hip/hip_runtime.h>
#include <hip/hip_bf16.h>
#include <math.h>

// ---------------------------------------------------------------------------
// OrderInvariantCapsuleLikelihood for MI455X (gfx1250, wave32, WMMA)
// B=32, N=1024, V=512, D=4.  Core pairwise term mapped onto
// V_WMMA_F32_16X16X4_F32 (K=4 == D).  Memory-bound (64MB softmax output).
// ---------------------------------------------------------------------------

#define BB 32
#define NN 1024
#define VV 512
#define DD 4
#define VP1 513
#define PSTRIDE 521   // LDS row stride; 521 mod 64 = 9 (coprime) -> no bank conflicts

#define EPS_C     1e-16f
#define LOG_2PI_C 1.8378770664093453f
#define DUMMY_C  (-4.605170185988091f)   // -2*ln(10)

// output layout (flat f32, reference tuple order)
#define O_LOGPROB    ((size_t)0)
#define O_VOTEPRES   ((size_t)1)                         // 32*512
#define O_WINNER     (O_VOTEPRES   + (size_t)BB*VV)      // 32*1024*4
#define O_WINNERPRES (O_WINNER     + (size_t)BB*NN*DD)   // 32*1024
#define O_ISFROM     (O_WINNERPRES + (size_t)BB*NN)      // 32*1024
#define O_MIXLOGITS  (O_ISFROM     + (size_t)BB*NN)      // 32*513
#define O_MIXLOGPROB (O_MIXLOGITS  + (size_t)BB*VP1)     // 32*513
#define O_ZERO_W     (O_MIXLOGPROB + (size_t)BB*VP1)     // 32*1024*4
#define O_ZERO_WP    (O_ZERO_W     + (size_t)BB*NN*DD)   // 32*1024
#define O_PMP        (O_ZERO_WP    + (size_t)BB*NN)      // 32*1024*512

typedef __attribute__((ext_vector_type(2))) float v2f;
typedef __attribute__((ext_vector_type(8))) float v8f;

// ---------------------------------------------------------------------------
// Kernel 1: mixing logits, log_softmax over V+1, vote_presence.  1 block / b.
// ---------------------------------------------------------------------------
__global__ void capsule_mixing_kernel(const float* __restrict__ vpp,
                                      float* __restrict__ out) {
  __shared__ float sdata[256];
  const int b   = blockIdx.x;
  const int tid = threadIdx.x;

  const float l0 = logf(vpp[(size_t)b * VV + tid]        + EPS_C);
  const float l1 = logf(vpp[(size_t)b * VV + tid + 256]  + EPS_C);

  // max over 513 logits (DUMMY folded in; max is idempotent)
  float m = fmaxf(fmaxf(l0, l1), DUMMY_C);
  sdata[tid] = m;  __syncthreads();
  for (int s = 128; s > 0; s >>= 1) {
    if (tid < s) sdata[tid] = fmaxf(sdata[tid], sdata[tid + s]);
    __syncthreads();
  }
  const float smax = sdata[0];  __syncthreads();

  float sum = expf(l0 - smax) + expf(l1 - smax);
  if (tid == 0) sum += expf(DUMMY_C - smax);
  sdata[tid] = sum;  __syncthreads();
  for (int s = 128; s > 0; s >>= 1) {
    if (tid < s) sdata[tid] += sdata[tid + s];
    __syncthreads();
  }
  const float logZ = smax + logf(sdata[0]);

  const size_t lb = (size_t)b * VP1;
  out[O_MIXLOGITS  + lb + tid]        = l0;
  out[O_MIXLOGITS  + lb + tid + 256]  = l1;
  out[O_MIXLOGPROB + lb + tid]        = l0 - logZ;
  out[O_MIXLOGPROB + lb + tid + 256]  = l1 - logZ;
  out[O_VOTEPRES + (size_t)b * VV + tid]       = (l0 > DUMMY_C) ? 1.0f : 0.0f;
  out[O_VOTEPRES + (size_t)b * VV + tid + 256] = (l1 > DUMMY_C) ? 1.0f : 0.0f;
  if (tid == 0) {
    out[O_MIXLOGITS  + lb + VV] = DUMMY_C;
    out[O_MIXLOGPROB + lb + VV] = DUMMY_C - logZ;
  }
}

// ---------------------------------------------------------------------------
// Kernel 2: zero-fill zeros_like outputs (d_out is poisoned by harness).
// ---------------------------------------------------------------------------
__global__ void capsule_zero_kernel(float* __restrict__ out) {
  const size_t n = (size_t)BB * NN * DD + (size_t)BB * NN; // 163840
  for (size_t i = blockIdx.x * (size_t)blockDim.x + threadIdx.x;
       i < n; i += (size_t)gridDim.x * blockDim.x)
    out[O_ZERO_W + i] = 0.0f;
}

// ---------------------------------------------------------------------------
// Kernel 3: main — one wave (32 threads) per (b, 16-row N tile).
// WMMA f32 16x16x4 computes posterior tiles; LDS stages 16x513 posterior.
// ---------------------------------------------------------------------------
__global__ void
__launch_bounds__(32)
capsule_main_kernel(const float* __restrict__ x,
                    const float* __restrict__ votes,
                    const float* __restrict__ scales,
                    const float* __restrict__ vpp,
                    const float* __restrict__ presence,
                    const int*   __restrict__ nvotes_p,
                    float* __restrict__ out,
                    float* __restrict__ partials) {
  __shared__ float post[16][PSTRIDE]; // posterior tile, cols 0..512 (512=dummy)
  __shared__ float xxs[16];           // ||x_row||^2
  __shared__ float lses[16];          // per-row logsumexp

  const int b    = blockIdx.x >> 6;          // 64 tiles per batch
  const int n0   = (blockIdx.x & 63) << 4;
  const int lane = threadIdx.x;
  const int half = lane >> 4;                // 0: K=0,1   1: K=2,3
  const int l15  = lane & 15;
  const int kb   = half * 2;

  // ---- A operand: x rows (M = l15), 2 of 4 dims per lane -------------------
  const float* xp = x + ((size_t)b * NN + n0 + l15) * DD + kb;
  const float a0 = xp[0], a1 = xp[1];
  v2f av; av.x = a0; av.y = a1;

  // full ||x||^2 per row via cross-half shuffle, staged in LDS
  const float xxp = a0 * a0 + a1 * a1;
  const float xx  = xxp + __shfl_xor(xxp, 16, 32);
  if (half == 0) xxs[l15] = xx;
  __syncthreads();
  const int rbase = half * 8;                // C/D rows owned by this lane
  float xr[8];
#pragma unroll
  for (int j = 0; j < 8; ++j) xr[j] = xxs[rbase + j];

  const float* mlp = out + O_MIXLOGPROB + (size_t)b * VP1;

  // ---- pass 1: 32 vote tiles of 16, one WMMA each --------------------------
  for (int vt = 0; vt < VV / 16; ++vt) {
    const int vcol = vt * 16 + l15;          // this lane's B column (N index)
    const float* vp = votes + ((size_t)b * VV + vcol) * DD + kb;
    const float m0 = vp[0], m1 = vp[1];
    const float sig  = scales[(size_t)b * VV + vcol];
    const float inv2 = 1.0f / (sig * sig);

    const float mmp = m0 * m0 + m1 * m1;
    const float mm  = mmp + __shfl_xor(mmp, 16, 32);
    const float cv  = -0.5f * inv2 * mm - 4.0f * logf(sig)
                      - 2.0f * LOG_2PI_C + mlp[vcol];

    v2f bv; bv.x = m0 * inv2; bv.y = m1 * inv2;   // fold 1/sigma^2 into B

    v8f acc;                                      // preload additive terms
#pragma unroll
    for (int j = 0; j < 8; ++j) acc[j] = cv - 0.5f * inv2 * xr[j];

    // posterior tile: D = A(16x4) * B(4x16) + C
    acc = __builtin_amdgcn_wmma_f32_16x16x4_f32(
        false, av, false, bv, (short)0, acc, false, false);

#pragma unroll
    for (int j = 0; j < 8; ++j) post[rbase + j][vcol] = acc[j];
  }
  // dummy column of posterior
  if (half == 0) post[l15][VV] = DUMMY_C + mlp[VV];
  __syncthreads();

  // ---- pass 2: per-row max/argmax/logsumexp (2 lanes per row) --------------
  const int r      = l15;
  const int cstart = half * 256;

  // argmax scan over 256 non-dummy columns (max of scanned == bestv)
  float bestv = -INFINITY;
  int   besti = 0;
#pragma unroll 4
  for (int c = cstart; c < cstart + 256; ++c) {
    const float p = post[r][c];
    if (p > bestv) { bestv = p; besti = c; }
  }
  float m = bestv;
  if (half) m = fmaxf(m, post[r][VV]);       // fold dummy column into max only
  const float rowmax = fmaxf(m, __shfl_xor(m, 16, 32));
  {
    const float obv = __shfl_xor(bestv, 16, 32);
    const int   obi = __shfl_xor(besti, 16, 32);
    if (obv > bestv || (obv == bestv && obi < besti)) { bestv = obv; besti = obi; }
  }
  float s = 0.0f;
#pragma unroll 4
  for (int c = cstart; c < cstart + 256; ++c) s += expf(post[r][c] - rowmax);
  if (half) s += expf(post[r][VV] - rowmax);
  s += __shfl_xor(s, 16, 32);
  const float lse = rowmax + logf(s);

  // weighted log-prob contribution (count each row once)
  const float pres = presence[(size_t)b * NN + n0 + r];
  float contrib = (half == 0) ? lse * pres : 0.0f;
#pragma unroll
  for (int off = 16; off > 0; off >>= 1) contrib += __shfl_xor(contrib, off, 32);
  if (lane == 0) partials[blockIdx.x] = contrib;

  // winner outputs
  if (half == 0) {
    const int nrow = n0 + r;
    const float* wv = votes + ((size_t)b * VV + besti) * DD;
    float* wo = out + O_WINNER + ((size_t)b * NN + nrow) * DD;
#pragma unroll
    for (int d = 0; d < DD; ++d) wo[d] = wv[d];
    out[O_WINNERPRES + (size_t)b * NN + nrow] = vpp[(size_t)b * VV + besti];
    out[O_ISFROM     + (size_t)b * NN + nrow] = (float)(besti / nvotes_p[0]);
    lses[r] = lse;
  }
  __syncthreads();

  // posterior_mixing_probs = softmax(posterior)[..., :512]
  // 64MB streaming output -> non-temporal stores (don't thrash L2)
  for (int r2 = 0; r2 < 16; ++r2) {
    const float l = lses[r2];
    float* pout = out + O_PMP + (((size_t)b * NN + n0 + r2) << 9);
#pragma unroll 4
    for (int c = lane; c < VV; c += 32)
      __builtin_nontemporal_store(expf(post[r2][c] - l), pout + c);
  }
}

// ---------------------------------------------------------------------------
// Kernel 4: deterministic fixed-order reduction of 2048 block partials.
// ---------------------------------------------------------------------------
__global__ void capsule_reduce_kernel(const float* __restrict__ partials,
                                      float* __restrict__ out) {
  __shared__ float sdata[256];
  const int tid = threadIdx.x;
  float s = 0.0f;
  for (int i = tid; i < BB * (NN / 16); i += 256) s += partials[i];
  sdata[tid] = s;  __syncthreads();
  for (int st = 128; st > 0; st >>= 1) {
    if (tid < st) sdata[tid] += sdata[tid + st];
    __syncthreads();
  }
  if (tid == 0) out[O_LOGPROB] = sdata[0];
}

// ---------------------------------------------------------------------------
extern "C" void kernel_launch(void* const* d_in, const int* in_sizes, int n_in,
                              void* d_out, int out_size, void* d_ws, size_t ws_size,
                              hipStream_t stream) {
  const float* x        = (const float*)d_in[0];
  const float* votes    = (const float*)d_in[1];
  const float* scales   = (const float*)d_in[2];
  const float* vpp      = (const float*)d_in[3];
  const float* presence = (const float*)d_in[4];
  const int*   nvotes   = (const int*)  d_in[5];
  float* out = (float*)d_out;
  float* ws  = (float*)d_ws;                 // 2048 partials (8 KB)

  capsule_mixing_kernel<<<BB, 256, 0, stream>>>(vpp, out);
  capsule_zero_kernel  <<<320, 256, 0, stream>>>(out);
  capsule_main_kernel  <<<BB * (NN / 16), 32, 0, stream>>>(
      x, votes, scales, vpp, presence, nvotes, out, ws);
  capsule_reduce_kernel<<<1, 256, 0, stream>>>(ws, out);
}